// MixLoraSparseMoe_45088566673916
// MI455X (gfx1250) — compile-verified
//
#include <hip/hip_runtime.h>
#include <hip/hip_bf16.h>

// ---------------------------------------------------------------------------
// MixLoRA sparse-MoE MLP for MI455X (gfx1250, wave32, WMMA).
// Sizes fixed per reference: N=1024 tokens, D=1024, F=4096, R=16, E=8, top-2.
// Big GEMMs on V_WMMA_F32_16X16X4_F32 (exact fp32 matrix math); A tiles are
// staged into LDS with GLOBAL_LOAD_ASYNC_TO_LDS_B128 (ASYNCcnt-tracked).
// ---------------------------------------------------------------------------

#define NTOK 1024
#define DD   1024
#define FF   4096
#define RR   16
#define EE   8
#define SCALE_F 2.0f
#define KC   128          // K-chunk staged in LDS
#define LDA  132          // padded LDS row stride (banks: 4*row mod 64 distinct)

typedef float v2f __attribute__((ext_vector_type(2)));
typedef float v8f __attribute__((ext_vector_type(8)));

__device__ __forceinline__ void async_ld_b128(unsigned lds_off,
                                              const float* gaddr) {
  asm volatile("global_load_async_to_lds_b128 %0, %1, off"
               :: "v"(lds_off), "v"(gaddr) : "memory");
}
__device__ __forceinline__ void wait_async0() {
  asm volatile("s_wait_asynccnt 0x0" ::: "memory");
}

// ---------------------------------------------------------------------------
// Kernel 1: router.  logits = x @ gate_w^T, softmax, top-2, renormalize,
// scatter into dense weight[N, E] (zeros for inactive experts).
// ---------------------------------------------------------------------------
__global__ __launch_bounds__(256) void router_kernel(
    const float* __restrict__ x, const float* __restrict__ gw,
    float* __restrict__ weight) {
  const int n = blockIdx.x * blockDim.x + threadIdx.x;
  if (n >= NTOK) return;

  float lg[EE];
#pragma unroll
  for (int e = 0; e < EE; ++e) lg[e] = 0.f;
  for (int d = 0; d < DD; ++d) {
    const float xv = x[n * DD + d];
#pragma unroll
    for (int e = 0; e < EE; ++e) lg[e] += xv * gw[e * DD + d];
  }
  float mx = lg[0];
#pragma unroll
  for (int e = 1; e < EE; ++e) mx = fmaxf(mx, lg[e]);
  float p[EE], s = 0.f;
#pragma unroll
  for (int e = 0; e < EE; ++e) { p[e] = __expf(lg[e] - mx); s += p[e]; }
  const float inv_s = 1.f / s;
#pragma unroll
  for (int e = 0; e < EE; ++e) p[e] *= inv_s;

  // top-2 (ties -> lowest index, matching top_k)
  int i0 = 0; float v0 = p[0];
#pragma unroll
  for (int e = 1; e < EE; ++e) if (p[e] > v0) { v0 = p[e]; i0 = e; }
  int i1 = -1; float v1 = -1.f;
#pragma unroll
  for (int e = 0; e < EE; ++e) if (e != i0 && p[e] > v1) { v1 = p[e]; i1 = e; }

  const float inv = 1.f / (v0 + v1);
#pragma unroll
  for (int e = 0; e < EE; ++e) weight[n * EE + e] = 0.f;
  weight[n * EE + i0] = v0 * inv;
  weight[n * EE + i1] = v1 * inv;
}

// ---------------------------------------------------------------------------
// Kernel 2: LoRA-A projections via WMMA.
//   t1[e,n,r] = sum_d x[n,d] * A1[e,r,d]   (and t3 with A3)
// One wave computes a 16(token)x16(rank) tile for both t1 and t3, K over D.
// grid = (NTOK/16/8, EE), block = 256 (8 waves).
// ---------------------------------------------------------------------------
__global__ __launch_bounds__(256) void gemm_loraA(
    const float* __restrict__ x, const float* __restrict__ A1,
    const float* __restrict__ A3, float* __restrict__ t1,
    float* __restrict__ t3) {
  const int lane = threadIdx.x & 31;
  const int wave = threadIdx.x >> 5;
  const int ntile = blockIdx.x * 8 + wave;   // 0..63
  const int e = blockIdx.y;                  // 0..7
  const int m = lane & 15;
  const int kh = (lane >> 4) << 1;           // 0 or 2
  const int arow = ntile * 16 + m;

  v8f c1 = {}, c3 = {};
  const float* a1p = A1 + (e * RR + m) * DD; // A1[e][r=m][.]
  const float* a3p = A3 + (e * RR + m) * DD;
  for (int k = 0; k < DD; k += 4) {
    const v2f a  = *(const v2f*)(x + arow * DD + k + kh);
    const v2f b1 = *(const v2f*)(a1p + k + kh);
    const v2f b3 = *(const v2f*)(a3p + k + kh);
    c1 = __builtin_amdgcn_wmma_f32_16x16x4_f32(false, a, false, b1, (short)0,
                                               c1, false, false);
    c3 = __builtin_amdgcn_wmma_f32_16x16x4_f32(false, a, false, b3, (short)0,
                                               c3, false, false);
  }
#pragma unroll
  for (int v = 0; v < 8; ++v) {
    const int row = ntile * 16 + v + ((lane >> 4) << 3);
    const int col = lane & 15;
    t1[(e * NTOK + row) * RR + col] = c1[v];
    t3[(e * NTOK + row) * RR + col] = c3[v];
  }
}

// ---------------------------------------------------------------------------
// Kernel 3: shared base gate/up GEMMs, fused (A fragment of x reused).
//   w1 = x @ w_gate^T, w3 = x @ w_up^T
// Block stages one 16x128 x-tile into LDS via async-to-LDS copies (one
// coalesced 512B row per wave), then 8 waves each drive 16 tok x 64 f cols
// for both outputs with A fragments read from LDS. grid = (64, 8).
// ---------------------------------------------------------------------------
__global__ __launch_bounds__(256) void gemm_gateup(
    const float* __restrict__ x, const float* __restrict__ wg,
    const float* __restrict__ wu, float* __restrict__ w1,
    float* __restrict__ w3) {
  __shared__ float tA[16][LDA];
  const int lane = threadIdx.x & 31;
  const int wave = threadIdx.x >> 5;
  const int ntile = blockIdx.x;                    // 0..63
  const int f0 = blockIdx.y * 512 + wave * 64;     // 4 tiles of 16
  const int m = lane & 15;
  const int kh = (lane >> 4) << 1;

  const int lrow = threadIdx.x >> 5;               // 0..7
  const int lcol = (threadIdx.x & 31) << 2;        // 0..124 (float4 lanes)

  v8f c1[4] = {}, c3[4] = {};
  for (int k0 = 0; k0 < DD; k0 += KC) {
    __syncthreads();
    {
      const float* g0 = x + (size_t)(ntile * 16 + lrow) * DD + k0 + lcol;
      const float* g1 = g0 + (size_t)8 * DD;
      async_ld_b128((unsigned)(unsigned long long)&tA[lrow][lcol], g0);
      async_ld_b128((unsigned)(unsigned long long)&tA[lrow + 8][lcol], g1);
      wait_async0();
    }
    __syncthreads();
    // prefetch next chunk of the streamed B rows
    if (k0 + KC < DD) {
#pragma unroll
      for (int t = 0; t < 4; ++t) {
        const int fcol = f0 + t * 16 + m;
        __builtin_prefetch(wg + (size_t)fcol * DD + k0 + KC + kh, 0, 1);
        __builtin_prefetch(wu + (size_t)fcol * DD + k0 + KC + kh, 0, 1);
      }
    }
#pragma unroll 8
    for (int kk = 0; kk < KC; kk += 4) {
      const v2f a = *(const v2f*)&tA[m][kk + kh];      // ds_load_b64
#pragma unroll
      for (int t = 0; t < 4; ++t) {
        const int fcol = f0 + t * 16 + m;
        const v2f b1 = *(const v2f*)(wg + (size_t)fcol * DD + k0 + kk + kh);
        const v2f b3 = *(const v2f*)(wu + (size_t)fcol * DD + k0 + kk + kh);
        c1[t] = __builtin_amdgcn_wmma_f32_16x16x4_f32(false, a, false, b1,
                                                      (short)0, c1[t], false, false);
        c3[t] = __builtin_amdgcn_wmma_f32_16x16x4_f32(false, a, false, b3,
                                                      (short)0, c3[t], false, false);
      }
    }
  }
#pragma unroll
  for (int t = 0; t < 4; ++t) {
#pragma unroll
    for (int v = 0; v < 8; ++v) {
      const int row = ntile * 16 + v + ((lane >> 4) << 3);
      const int col = f0 + t * 16 + (lane & 15);
      w1[row * FF + col] = c1[t][v];
      w3[row * FF + col] = c3[t][v];
    }
  }
}

// ---------------------------------------------------------------------------
// Kernel 4: fused per-token epilogue (only the 2 live experts do work).
//   d1 = SCALE * t1[e,n,:] . B1[e,f,:]    (rank-16)
//   act = silu(w1 + d1) * (w3 + d3)
//   actsum[n,f] = sum_e weight[n,e] * act
//   zw[n, e*16+r] = weight[n,e] * SCALE * sum_f act * A2[e,r,f]
// Block per token, 256 threads, each owns 16 f's (stride-256, coalesced).
// z reduction: wave shfl_xor then LDS across the 8 waves.
// ---------------------------------------------------------------------------
__global__ __launch_bounds__(256) void act_kernel(
    const float* __restrict__ w1, const float* __restrict__ w3,
    const float* __restrict__ t1, const float* __restrict__ t3,
    const float* __restrict__ weight, const float* __restrict__ B1,
    const float* __restrict__ B3, const float* __restrict__ A2,
    float* __restrict__ actsum, float* __restrict__ zw) {
  const int n = blockIdx.x;
  const int tid = threadIdx.x;
  const int lane = tid & 31;
  const int wave = tid >> 5;

  __shared__ float wrow[EE];
  __shared__ float t1s[EE][RR];
  __shared__ float t3s[EE][RR];
  __shared__ float zpart[8][RR];

  if (tid < EE) wrow[tid] = weight[n * EE + tid];
  if (tid < EE * RR) {
    t1s[tid >> 4][tid & 15] = t1[((tid >> 4) * NTOK + n) * RR + (tid & 15)];
  } else if (tid < 2 * EE * RR) {
    const int q = tid - EE * RR;
    t3s[q >> 4][q & 15] = t3[((q >> 4) * NTOK + n) * RR + (q & 15)];
  }
  __syncthreads();

  float acc[16];
#pragma unroll
  for (int j = 0; j < 16; ++j) acc[j] = 0.f;

  for (int e = 0; e < EE; ++e) {
    const float w = wrow[e];             // uniform across the block
    if (w == 0.f) {                      // inactive expert: still zero zw
      if (tid < RR) zw[n * (EE * RR) + e * RR + tid] = 0.f;
      continue;
    }
    float zacc[16];
#pragma unroll
    for (int r = 0; r < 16; ++r) zacc[r] = 0.f;

#pragma unroll 4
    for (int j = 0; j < 16; ++j) {
      const int f = tid + (j << 8);
      const float* b1p = B1 + (size_t)(e * FF + f) * RR;
      const float* b3p = B3 + (size_t)(e * FF + f) * RR;
      float d1 = 0.f, d3 = 0.f;
#pragma unroll
      for (int r = 0; r < 16; ++r) {
        d1 += t1s[e][r] * b1p[r];
        d3 += t3s[e][r] * b3p[r];
      }
      const float v1 = w1[n * FF + f] + SCALE_F * d1;
      const float v3 = w3[n * FF + f] + SCALE_F * d3;
      const float a = (v1 / (1.f + __expf(-v1))) * v3;  // silu(v1)*v3
      acc[j] += w * a;
      const float* a2p = A2 + (size_t)e * RR * FF + f;
#pragma unroll
      for (int r = 0; r < 16; ++r) zacc[r] += a * a2p[r * FF];
    }
    // wave-level butterfly reduce: every lane ends with full sums
#pragma unroll
    for (int r = 0; r < 16; ++r) {
      zacc[r] += __shfl_xor(zacc[r], 16, 32);
      zacc[r] += __shfl_xor(zacc[r], 8, 32);
      zacc[r] += __shfl_xor(zacc[r], 4, 32);
      zacc[r] += __shfl_xor(zacc[r], 2, 32);
      zacc[r] += __shfl_xor(zacc[r], 1, 32);
    }
    float mine = 0.f;                    // lane l (l<16) picks zacc[l]
#pragma unroll
    for (int r = 0; r < 16; ++r) if (lane == r) mine = zacc[r];
    if (lane < RR) zpart[wave][lane] = mine;
    __syncthreads();
    if (tid < RR) {
      float s = 0.f;
#pragma unroll
      for (int wv = 0; wv < 8; ++wv) s += zpart[wv][tid];
      zw[n * (EE * RR) + e * RR + tid] = w * SCALE_F * s;
    }
    __syncthreads();
  }
#pragma unroll
  for (int j = 0; j < 16; ++j) actsum[n * FF + tid + (j << 8)] = acc[j];
}

// ---------------------------------------------------------------------------
// Kernel 5: down projection + LoRA-down, one fused WMMA accumulator:
//   out = actsum @ w_down^T  (K = F = 4096, A tile LDS-staged via async copy)
//       + zw     @ B2flat^T  (K = E*R = 128, weight & SCALE pre-folded)
// B2flat[d, e*16+r] = B2[e, d, r].  grid = (64, 2), block = 256 (8 waves).
// ---------------------------------------------------------------------------
__global__ __launch_bounds__(256) void gemm_down(
    const float* __restrict__ acts, const float* __restrict__ wd,
    const float* __restrict__ zw, const float* __restrict__ B2,
    float* __restrict__ out) {
  __shared__ float tA[16][LDA];
  const int lane = threadIdx.x & 31;
  const int wave = threadIdx.x >> 5;
  const int ntile = blockIdx.x;                 // 0..63
  const int d0 = blockIdx.y * 512 + wave * 64;  // 4 tiles of 16
  const int m = lane & 15;
  const int kh = (lane >> 4) << 1;
  const int arow = ntile * 16 + m;

  const int lrow = threadIdx.x >> 5;            // 0..7
  const int lcol = (threadIdx.x & 31) << 2;     // 0..124

  v8f c[4] = {};
  for (int k0 = 0; k0 < FF; k0 += KC) {
    __syncthreads();
    {
      const float* g0 = acts + (size_t)(ntile * 16 + lrow) * FF + k0 + lcol;
      const float* g1 = g0 + (size_t)8 * FF;
      async_ld_b128((unsigned)(unsigned long long)&tA[lrow][lcol], g0);
      async_ld_b128((unsigned)(unsigned long long)&tA[lrow + 8][lcol], g1);
      wait_async0();
    }
    __syncthreads();
#pragma unroll 8
    for (int kk = 0; kk < KC; kk += 4) {
      const v2f a = *(const v2f*)&tA[m][kk + kh];      // ds_load_b64
#pragma unroll
      for (int t = 0; t < 4; ++t) {
        const int dcol = d0 + t * 16 + m;
        const v2f b = *(const v2f*)(wd + (size_t)dcol * FF + k0 + kk + kh);
        c[t] = __builtin_amdgcn_wmma_f32_16x16x4_f32(false, a, false, b,
                                                     (short)0, c[t], false, false);
      }
    }
  }
  // LoRA-down tail: K = 128, pairs (kk, kk+1) never cross an expert boundary
  for (int k = 0; k < EE * RR; k += 4) {
    const int kk = k + kh;                 // even
    const int e = kk >> 4;
    const int r = kk & 15;
    const v2f a = *(const v2f*)(zw + arow * (EE * RR) + kk);
#pragma unroll
    for (int t = 0; t < 4; ++t) {
      const int dcol = d0 + t * 16 + m;
      const v2f b = *(const v2f*)(B2 + ((size_t)e * DD + dcol) * RR + r);
      c[t] = __builtin_amdgcn_wmma_f32_16x16x4_f32(false, a, false, b,
                                                   (short)0, c[t], false, false);
    }
  }
#pragma unroll
  for (int t = 0; t < 4; ++t) {
#pragma unroll
    for (int v = 0; v < 8; ++v) {
      const int row = ntile * 16 + v + ((lane >> 4) << 3);
      const int col = d0 + t * 16 + (lane & 15);
      out[row * DD + col] = c[t][v];
    }
  }
}

// ---------------------------------------------------------------------------
// Host-side launch: stage everything through d_ws (≈52 MB of fp32).
// in order: hidden, gate_w, w_gate, w_up, w_down, A1, B1, A3, B3, A2, B2
// ---------------------------------------------------------------------------
extern "C" void kernel_launch(void* const* d_in, const int* in_sizes, int n_in,
                              void* d_out, int out_size, void* d_ws,
                              size_t ws_size, hipStream_t stream) {
  const float* x      = (const float*)d_in[0];
  const float* gate_w = (const float*)d_in[1];
  const float* w_gate = (const float*)d_in[2];
  const float* w_up   = (const float*)d_in[3];
  const float* w_down = (const float*)d_in[4];
  const float* A1     = (const float*)d_in[5];
  const float* B1     = (const float*)d_in[6];
  const float* A3     = (const float*)d_in[7];
  const float* B3     = (const float*)d_in[8];
  const float* A2     = (const float*)d_in[9];
  const float* B2     = (const float*)d_in[10];
  float* out = (float*)d_out;

  float* ws     = (float*)d_ws;
  float* weight = ws;                       // N*E      =     8192
  float* t1     = weight + NTOK * EE;       // E*N*R    =   131072
  float* t3     = t1 + EE * NTOK * RR;      // E*N*R    =   131072
  float* w1     = t3 + EE * NTOK * RR;      // N*F      =  4194304
  float* w3     = w1 + (size_t)NTOK * FF;   // N*F      =  4194304
  float* acts   = w3 + (size_t)NTOK * FF;   // N*F      =  4194304
  float* zw     = acts + (size_t)NTOK * FF; // N*E*R    =   131072

  router_kernel<<<NTOK / 256, 256, 0, stream>>>(x, gate_w, weight);
  gemm_loraA<<<dim3(NTOK / 16 / 8, EE), 256, 0, stream>>>(x, A1, A3, t1, t3);
  gemm_gateup<<<dim3(NTOK / 16, FF / 512), 256, 0, stream>>>(x, w_gate, w_up,
                                                             w1, w3);
  act_kernel<<<NTOK, 256, 0, stream>>>(w1, w3, t1, t3, weight, B1, B3, A2,
                                       acts, zw);
  gemm_down<<<dim3(NTOK / 16, DD / 512), 256, 0, stream>>>(acts, w_down, zw,
                                                           B2, out);
}